// CNNDSTv2_optimized_20796231647894
// MI455X (gfx1250) — compile-verified
//
#include <hip/hip_runtime.h>

// ---------------------------------------------------------------------------
// Dempster combination via fast zeta / Moebius transforms over the 2^24
// subset lattice.  Each 4-bit "digit" transform is a constant 16x16 matrix
// multiply executed as 4 chained V_WMMA_F32_16X16X4_F32 (CDNA5 wave32 WMMA).
// 3-pass structure (576 MB total traffic, memory-bound):
//   P1 k_low4  : zeta d0..d3 on contiguous 64K-float tiles (266KB dyn LDS)
//   P2 k_mid2  : TDM tile loads -> zeta d4,d5 (both ch) + product + Moebius
//   P3 k_low4  : Moebius d3..d0 in place
// ---------------------------------------------------------------------------

typedef __attribute__((ext_vector_type(2))) float        v2f;
typedef __attribute__((ext_vector_type(8))) float        v8f;
typedef __attribute__((ext_vector_type(4))) unsigned int v4u;
typedef __attribute__((ext_vector_type(8))) int          v8i;
typedef __attribute__((ext_vector_type(4))) int          v4i;

static constexpr size_t LFRAME = (size_t)1 << 24;   // 16,777,216

// LDS skew padding: +1 float per 64 -> kills 64-float-stride bank conflicts.
// (Matches TDM pad_enable=1, pad_interval=5 (64 DWORDs), pad_amount=0 (+1).)
#define LPAD(i) ((i) + ((i) >> 6))

extern __shared__ float smem[];

#if __has_builtin(__builtin_amdgcn_tensor_load_to_lds) && \
    __has_builtin(__builtin_amdgcn_s_wait_tensorcnt)
#define HAVE_TDM 1
#else
#define HAVE_TDM 0
#endif

#if HAVE_TDM
// One 2D TDM descriptor: tile_d1 rows of tile_d0 4-byte elements, global row
// stride row_stride elements, packed into LDS at lds_byte with +1 DWORD pad
// after every 64 DWORDs stored (== LPAD layout).  tensor dims == tile dims.
// clang-23 builtin: (v4u g0, v8i g1, v4i g2, v4i g3, v8i vaddr4(unused), i32 cpol)
__device__ __forceinline__ void tdm_load_2d(const void* gptr, unsigned lds_byte,
                                            unsigned tile_d0, unsigned tile_d1,
                                            unsigned long long row_stride) {
  const unsigned long long ga = (unsigned long long)(size_t)gptr;
  v4u g0;
  g0.x = 1u;                                        // count=1, user descriptor
  g0.y = lds_byte;                                  // lds_addr (bytes)
  g0.z = (unsigned)(ga & 0xffffffffu);              // global_addr[31:0]
  g0.w = (unsigned)((ga >> 32) & 0x01ffffffu)       // global_addr[56:32]
       | (2u << 30);                                // type = 2 ("image")
  v8i g1;
  g1[0] = (int)((2u << 16)                          // data_size = 4 bytes
              | (1u << 20)                          // pad_enable
              | (5u << 22));                        // pad_interval=64 DW, amt=1
  g1[1] = (int)((tile_d0 & 0xffffu) << 16);         // tensor_dim0[15:0]
  g1[2] = (int)(((tile_d0 >> 16) & 0xffffu)         // tensor_dim0[31:16]
              | ((tile_d1 & 0xffffu) << 16));       // tensor_dim1[15:0]
  g1[3] = (int)(((tile_d1 >> 16) & 0xffffu)         // tensor_dim1[31:16]
              | ((tile_d0 & 0xffffu) << 16));       // tile_dim0
  g1[4] = (int)(tile_d1 & 0xffffu);                 // tile_dim1 (tile_dim2=0)
  g1[5] = (int)(unsigned)(row_stride & 0xffffffffu);        // dim0_stride lo
  g1[6] = (int)(unsigned)((row_stride >> 32) & 0xffffu);    // dim0_stride hi
  g1[7] = 0;                                        // dim1_stride = 0 (2D)
  const v4i z4 = {0, 0, 0, 0};                      // groups 2/3 unused (2D)
  const v8i z8 = {0, 0, 0, 0, 0, 0, 0, 0};          // VADDR4: unused per ISA
  __builtin_amdgcn_tensor_load_to_lds(g0, g1, z4, z4, z8, 0);
}
#endif

// M[i][j] for zeta (sum over supersets) / Moebius (signed inverse) on 4 bits.
__device__ __forceinline__ float matval(int i, int j, bool mob) {
  if ((i & j) != i) return 0.0f;                 // need j superset of i
  if (!mob) return 1.0f;
  return (__popc(i ^ j) & 1) ? -1.0f : 1.0f;     // (-1)^{|j \ i|}
}

// Apply the 16x16 digit transform along stride 2^ls inside an LDS tile.
// One 16x16 WMMA macro-tile = 16 vectors; 4 chained 16x16x4 f32 WMMAs.
template <bool MOB, int NTHR>
__device__ __forceinline__ void digit_stage(float* lds, int nvec, int ls) {
  const int lane = threadIdx.x & 31;
  const int wave = threadIdx.x >> 5;
  const int half = lane >> 4;        // K/M-half select per ISA wave32 layouts
  const int col  = lane & 15;        // A row index / B,D column index

  v2f A[4];                          // A = M cols [4k+2*half, +1] (16x4 layout)
#pragma unroll
  for (int k = 0; k < 4; ++k) {
    const int c0 = 4 * k + 2 * half;
    A[k].x = matval(col, c0, MOB);
    A[k].y = matval(col, c0 + 1, MOB);
  }

  const int s   = 1 << ls;
  const int msk = s - 1;
  const int nmt = nvec >> 4;                     // 16 vectors per macro-tile
  for (int mt = wave; mt < nmt; mt += (NTHR / 32)) {
    const int u    = (mt << 4) + col;
    const int base = ((u >> ls) << (ls + 4)) + (u & msk);
    v8f acc = {0.f, 0.f, 0.f, 0.f, 0.f, 0.f, 0.f, 0.f};
#pragma unroll
    for (int k = 0; k < 4; ++k) {                // chain K=4 slabs -> K=16
      const int r0 = 4 * k + 2 * half;           // B rows per wave32 layout
      v2f b;
      b.x = lds[LPAD(base + (r0    ) * s)];
      b.y = lds[LPAD(base + (r0 + 1) * s)];
      acc = __builtin_amdgcn_wmma_f32_16x16x4_f32(
          false, A[k], false, b, (short)0, acc, false, false);
    }
    const int rb = half * 8;                     // D rows: v / v+8 halves
#pragma unroll
    for (int v = 0; v < 8; ++v)
      lds[LPAD(base + (rb + v) * s)] = acc[v];
  }
}

// 4-bit butterfly on 16 contiguous elements held in registers (digit 0).
template <bool MOB>
__device__ __forceinline__ void rad16_regs(float x[16]) {
#pragma unroll
  for (int b = 1; b < 16; b <<= 1)
#pragma unroll
    for (int i = 0; i < 16; ++i)
      if (!(i & b)) x[i] = MOB ? (x[i] - x[i | b]) : (x[i] + x[i | b]);
}

// ---- P1/P3: digits d0 (regs), d1,d2,d3 (WMMA) on 65536-float tiles -------
// Dynamic LDS: (LPAD(65535)+1)*4 = 266,236 B  (<= 320KB CDNA5 WGP LDS)
template <bool MOB>
__global__ __launch_bounds__(1024) void k_low4(const float* __restrict__ src,
                                               float* __restrict__ dst) {
  float* lds = smem;
  const size_t tile = (size_t)blockIdx.x * 65536;
  const int t = threadIdx.x;

  const float4* s4 = (const float4*)(src + tile);
#pragma unroll
  for (int k = 0; k < 16; ++k) {                 // coalesced 128b loads
    const int e = t + k * 1024;                  // float4 id 0..16383
    const float4 d = s4[e];
    const int a = e * 4;
    lds[LPAD(a + 0)] = d.x; lds[LPAD(a + 1)] = d.y;
    lds[LPAD(a + 2)] = d.z; lds[LPAD(a + 3)] = d.w;
  }
  __syncthreads();

#pragma unroll
  for (int i = 0; i < 4; ++i) {                  // digit 0 in registers
    const int g = t + i * 1024;                  // 16-group id 0..4095
    float x[16];
#pragma unroll
    for (int v = 0; v < 16; ++v) x[v] = lds[LPAD(g * 16 + v)];
    rad16_regs<MOB>(x);
#pragma unroll
    for (int v = 0; v < 16; ++v) lds[LPAD(g * 16 + v)] = x[v];
  }
  __syncthreads();
  digit_stage<MOB, 1024>(lds, 4096, 4);          // digit 1, stride 16
  __syncthreads();
  digit_stage<MOB, 1024>(lds, 4096, 8);          // digit 2, stride 256
  __syncthreads();
  digit_stage<MOB, 1024>(lds, 4096, 12);         // digit 3, stride 4096
  __syncthreads();

  float4* d4p = (float4*)(dst + tile);
#pragma unroll
  for (int k = 0; k < 16; ++k) {
    const int e = t + k * 1024;
    const int a = e * 4;
    float4 d;
    d.x = lds[LPAD(a + 0)]; d.y = lds[LPAD(a + 1)];
    d.z = lds[LPAD(a + 2)]; d.w = lds[LPAD(a + 3)];
    d4p[e] = d;
  }
}

// ---- P2: zeta d4,d5 (both ch) + Dempster product + Moebius d5,d4 ---------
// Tile: 256 rows (r = d5*16+d4, global row stride 65536) x 32 contiguous.
// Loaded by TDM (one descriptor per channel, issued by wave 0) with the pad
// feature reproducing the LPAD skew.  Dynamic LDS: 2 x 8320 floats = 66,560 B.
__global__ __launch_bounds__(256) void k_mid2(float* __restrict__ q1,
                                              const float* __restrict__ q2) {
  float* la = smem;                              // q1 tile at LDS byte 0
  float* lb = smem + 8320;                       // q2 tile at LDS byte 33280
  const size_t rem = (size_t)blockIdx.x * 32;    // window into d0..d3 space
  const int t = threadIdx.x;

#if HAVE_TDM
  if ((t >> 5) == 0) {                           // wave 0 issues both DMAs
    tdm_load_2d(q1 + rem, 0u,     32u, 256u, 65536ull);
    tdm_load_2d(q2 + rem, 33280u, 32u, 256u, 65536ull);
    __builtin_amdgcn_s_wait_tensorcnt(0);        // TENSORcnt -> data in LDS
  }
  __syncthreads();                               // publish to all waves
#else
#pragma unroll
  for (int k = 0; k < 8; ++k) {                  // 128B segments per row
    const int e  = t + k * 256;                  // float4 id 0..2047
    const int r  = e >> 3;
    const int c4 = e & 7;
    const size_t g = (size_t)r * 65536 + rem + (size_t)c4 * 4;
    const float4 x = *(const float4*)(q1 + g);
    const float4 y = *(const float4*)(q2 + g);
    const int a = r * 32 + c4 * 4;
    la[LPAD(a + 0)] = x.x; la[LPAD(a + 1)] = x.y;
    la[LPAD(a + 2)] = x.z; la[LPAD(a + 3)] = x.w;
    lb[LPAD(a + 0)] = y.x; lb[LPAD(a + 1)] = y.y;
    lb[LPAD(a + 2)] = y.z; lb[LPAD(a + 3)] = y.w;
  }
  __syncthreads();
#endif

  digit_stage<false, 256>(la, 512, 5);           // zeta d4 (tile stride 32)
  digit_stage<false, 256>(lb, 512, 5);
  __syncthreads();
  digit_stage<false, 256>(la, 512, 9);           // zeta d5 (tile stride 512)
  digit_stage<false, 256>(lb, 512, 9);
  __syncthreads();
  for (int e = t; e < 8192; e += 256)            // Dempster product
    la[LPAD(e)] *= lb[LPAD(e)];
  __syncthreads();
  digit_stage<true, 256>(la, 512, 9);            // Moebius d5
  __syncthreads();
  digit_stage<true, 256>(la, 512, 5);            // Moebius d4
  __syncthreads();

#pragma unroll
  for (int k = 0; k < 8; ++k) {                  // VMEM store (de-padded)
    const int e  = t + k * 256;
    const int r  = e >> 3;
    const int c4 = e & 7;
    const size_t g = (size_t)r * 65536 + rem + (size_t)c4 * 4;
    const int a = r * 32 + c4 * 4;
    float4 d;
    d.x = la[LPAD(a + 0)]; d.y = la[LPAD(a + 1)];
    d.z = la[LPAD(a + 2)]; d.w = la[LPAD(a + 3)];
    *(float4*)(q1 + g) = d;
  }
}

// ---------------------------------------------------------------------------
extern "C" void kernel_launch(void* const* d_in, const int* in_sizes, int n_in,
                              void* d_out, int out_size, void* d_ws, size_t ws_size,
                              hipStream_t stream) {
  (void)in_sizes; (void)n_in; (void)out_size; (void)ws_size;
  const float* m12 = (const float*)d_in[0];      // (1, 2, 2^24) f32
  float* q1 = (float*)d_out;                     // channel-0 scratch / result
  float* q2 = (float*)d_ws;                      // channel-1 scratch (64 MB)

  const size_t LDS_LOW = (size_t)(LPAD(65535) + 1) * sizeof(float); // 266,236 B
  const size_t LDS_MID = (size_t)(2 * 8320) * sizeof(float);        //  66,560 B

  // P1: zeta on digits d0..d3, contiguous tiles, both channels
  k_low4<false><<<256, 1024, LDS_LOW, stream>>>(m12, q1);
  k_low4<false><<<256, 1024, LDS_LOW, stream>>>(m12 + LFRAME, q2);
  // P2: TDM tile loads, zeta d4,d5 + product + Moebius d5,d4 -> q1
  k_mid2<<<2048, 256, LDS_MID, stream>>>(q1, q2);
  // P3: Moebius on digits d3..d0, in place on q1 (= d_out)
  k_low4<true><<<256, 1024, LDS_LOW, stream>>>(q1, q1);
}